// NGCFLayer_17343077941929
// MI455X (gfx1250) — compile-verified
//
#include <hip/hip_runtime.h>
#include <hip/hip_bf16.h>

typedef __attribute__((ext_vector_type(2))) float v2f;
typedef __attribute__((ext_vector_type(8))) float v8f;

#define EMB_D 128

// ---------------------------------------------------------------------------
// Phase 0a: zero the segment-sum accumulator (d_ws is poisoned, not zeroed).
// ---------------------------------------------------------------------------
__global__ void ngcf_zero(float4* __restrict__ p, long long n4) {
    long long i = (long long)blockIdx.x * blockDim.x + threadIdx.x;
    if (i < n4) p[i] = make_float4(0.f, 0.f, 0.f, 0.f);
}

// ---------------------------------------------------------------------------
// Phase 0b: transpose the two 128x128 weight matrices into workspace so the
// GEMM's B-fragment K-pairs (W[k][n], W[k+1][n]) become contiguous and merge
// into global_load_b64.  64 KB each; stays L2-resident across all row tiles.
// ---------------------------------------------------------------------------
__global__ void ngcf_transpose2(const float* __restrict__ W1,
                                const float* __restrict__ W2,
                                float* __restrict__ W1t,
                                float* __restrict__ W2t) {
    for (int i = threadIdx.x + blockIdx.x * blockDim.x;
         i < EMB_D * EMB_D; i += blockDim.x * gridDim.x) {
        const int k = i >> 7;        // row of W  (K index)
        const int n = i & (EMB_D-1); // col of W  (N index)
        W1t[n * EMB_D + k] = W1[i];
        W2t[n * EMB_D + k] = W2[i];
    }
}

// ---------------------------------------------------------------------------
// Phase 1: SpMM scatter.  One wave32 per edge; lane L owns dims [4L, 4L+4).
// emb gather per edge = one contiguous 512B line set (global_load_b128 x32),
// scatter = 4 hardware f32 atomics per lane into the L2-resident accumulator.
// ---------------------------------------------------------------------------
__global__ __launch_bounds__(256) void ngcf_spmm(
    const float* __restrict__ emb,
    const int*   __restrict__ erow,
    const int*   __restrict__ ecol,
    const float* __restrict__ eval,
    float*       __restrict__ rel,
    int nEdges)
{
    const int wave = (blockIdx.x * (blockDim.x >> 5)) + (threadIdx.x >> 5);
    const int lane = threadIdx.x & 31;
    if (wave >= nEdges) return;

    const int   row = erow[wave];
    const int   col = ecol[wave];
    const float v   = eval[wave];

    const float4 e = *(const float4*)(emb + (long long)col * EMB_D + lane * 4);
    float* dst = rel + (long long)row * EMB_D + lane * 4;

    unsafeAtomicAdd(dst + 0, e.x * v);
    unsafeAtomicAdd(dst + 1, e.y * v);
    unsafeAtomicAdd(dst + 2, e.z * v);
    unsafeAtomicAdd(dst + 3, e.w * v);
}

// ---------------------------------------------------------------------------
// Phase 2: out = rel @ W1 + (rel .* emb) @ W2, fused into one f32 WMMA
// accumulator.  Workgroup = 256 threads = 8 waves = one 16-row tile; wave w
// computes the 16x16 tile at columns [16w, 16w+16).  K stepped by 4 using
// V_WMMA_F32_16X16X4_F32 (exact f32 semantics, matches the f32 reference).
//
// A 16x4 f32 layout (ISA 7.12.2): lane = M (0..15 repeated), VGPR0/1 = K
// pair, lanes 16..31 hold K+2.  B 4x16 mirrors it with lane = N (loaded from
// the pre-transposed weights so each K-pair is one b64).  C/D 16x16:
// VGPR i -> row M = i + 8*(lane>=16), lane&15 -> N.
// ---------------------------------------------------------------------------
__global__ __launch_bounds__(256) void ngcf_gemm(
    const float* __restrict__ rel,    // [N,128]
    const float* __restrict__ emb,    // [N,128]
    const float* __restrict__ W1t,    // [128,128] = W1 transposed (N-major)
    const float* __restrict__ W2t,    // [128,128] = W2 transposed (N-major)
    float*       __restrict__ out,    // [N,128]
    int nRows)
{
    const int row0  = blockIdx.x * 16;
    const int wave  = threadIdx.x >> 5;     // 0..7 -> column tile
    const int lane  = threadIdx.x & 31;
    const int col0  = wave * 16;
    const int m     = lane & 15;            // M for A-frag, N for B/C-frag
    const int khalf = (lane >> 4) << 1;     // 0 or 2 extra K for hi half-wave

    int r = row0 + m;
    if (r >= nRows) r = nRows - 1;          // branchless clamp (keeps EXEC full)
    const float* relRow = rel + (long long)r * EMB_D;
    const float* embRow = emb + (long long)r * EMB_D;
    const float* w1Row  = W1t + (long long)(col0 + m) * EMB_D;  // column nc of W1
    const float* w2Row  = W2t + (long long)(col0 + m) * EMB_D;  // column nc of W2

    v8f c = {};
#pragma unroll 4
    for (int k = 0; k < EMB_D; k += 4) {
        const int ka = k + khalf;

        v2f a1, a2, b1, b2;
        a1.x = relRow[ka];
        a1.y = relRow[ka + 1];
        a2.x = a1.x * embRow[ka];
        a2.y = a1.y * embRow[ka + 1];
        b1.x = w1Row[ka];
        b1.y = w1Row[ka + 1];
        b2.x = w2Row[ka];
        b2.y = w2Row[ka + 1];

        c = __builtin_amdgcn_wmma_f32_16x16x4_f32(
                false, a1, false, b1, (short)0, c, false, false);
        c = __builtin_amdgcn_wmma_f32_16x16x4_f32(
                false, a2, false, b2, (short)0, c, false, false);
    }

    const int mo = (lane >> 4) << 3;        // 0 or 8
    if (row0 + 16 <= nRows) {
        // Fast path: full tile, uniform branch, 8 coalesced unguarded stores.
#pragma unroll
        for (int i = 0; i < 8; ++i)
            out[(long long)(row0 + mo + i) * EMB_D + col0 + m] = c[i];
    } else {
        // Tail tile only (not hit when nRows % 16 == 0).
#pragma unroll
        for (int i = 0; i < 8; ++i) {
            const int rr = row0 + mo + i;
            if (rr < nRows)
                out[(long long)rr * EMB_D + col0 + m] = c[i];
        }
    }
}

// ---------------------------------------------------------------------------
// Host-side launcher.
// Inputs: [0] embeddings f32 [N*128], [1] edge_row i32 [E], [2] edge_col i32
// [E], [3] edge_val f32 [E], [4] W1 f32 [128*128], [5] W2 f32 [128*128].
// Output: f32 [N*128].
// Workspace layout: [0, N*128) f32 accumulator | two 128x128 f32 transposed
// weight matrices.
// ---------------------------------------------------------------------------
extern "C" void kernel_launch(void* const* d_in, const int* in_sizes, int n_in,
                              void* d_out, int out_size, void* d_ws, size_t ws_size,
                              hipStream_t stream) {
    const float* emb  = (const float*)d_in[0];
    const int*   erow = (const int*)  d_in[1];
    const int*   ecol = (const int*)  d_in[2];
    const float* eval = (const float*)d_in[3];
    const float* W1   = (const float*)d_in[4];
    const float* W2   = (const float*)d_in[5];
    float*       out  = (float*)d_out;

    const int N = in_sizes[0] / EMB_D;
    const int E = in_sizes[1];

    float* rel = (float*)d_ws;
    float* W1t = rel + (long long)N * EMB_D;
    float* W2t = W1t + EMB_D * EMB_D;

    // Phase 0a: zero accumulator (N*128 floats = N*32 float4s)
    const long long n4 = (long long)N * (EMB_D / 4);
    ngcf_zero<<<(unsigned)((n4 + 255) / 256), 256, 0, stream>>>((float4*)rel, n4);

    // Phase 0b: transpose weights (tiny; overlaps nothing but costs ~nothing)
    ngcf_transpose2<<<64, 256, 0, stream>>>(W1, W2, W1t, W2t);

    // Phase 1: wave-per-edge gather/scatter (8 waves per 256-thread block)
    const int edgeBlocks = (E + 7) / 8;
    ngcf_spmm<<<edgeBlocks, 256, 0, stream>>>(emb, erow, ecol, eval, rel, E);

    // Phase 2: fused dual-GEMM via f32 WMMA (one 16-row tile per block)
    const int tiles = (N + 15) / 16;
    ngcf_gemm<<<tiles, 256, 0, stream>>>(rel, emb, W1t, W2t, out, N);
}